// RoIBBox_41755672052246
// MI455X (gfx1250) — compile-verified
//
#include <hip/hip_runtime.h>

// ---------------------------------------------------------------------------
// RoIBBox proposal generation for MI455X (gfx1250, wave32).
// B=32 images, A=147456 anchors (128x128x9), top-6000 -> NMS(0.7) -> 300 ROIs.
//
// Phases:
//   k_init     : zero hist/counters, cand scores = -inf
//   k_softmax  : softmax over K=9 logits per cell          (HBM-bound, 1 pass)
//   k_hist     : 4096-bin LDS-atomic histogram of scores   (1 read pass)
//   k_thresh   : per-image threshold bin + boundary budget (exact top-6000)
//   k_compact  : compact top-6000, decode boxes ONLY for selected anchors
//   k_nms      : 1 workgroup/image, full candidate set in 144KB LDS
//                (possible because CDNA5 WGP has 320KB LDS), filled with
//                global_load_async_to_lds_b128 + s_wait_asynccnt,
//                wave32 shuffle argmax, 300 sequential suppress rounds.
// ---------------------------------------------------------------------------

static constexpr int KANCH = 9;
static constexpr int PRE   = 6000;
static constexpr int POST  = 300;
static constexpr int NBINS = 4096;
static constexpr float IOU_T = 0.7f;

#define NEG_INF (-__builtin_inff())

// CDNA5 async DMA: copy 16B from per-lane global address to per-lane LDS
// address, tracked with ASYNCcnt (not LOADcnt). lds_addr is the 32-bit LDS
// byte address (low 32 bits of the generic pointer — the flat aperture maps
// LDS by truncating to addr[31:0]).
__device__ __forceinline__ void async_b128_to_lds(unsigned lds_addr, const void* g) {
  asm volatile("global_load_async_to_lds_b128 %0, %1, off"
               :: "v"(lds_addr),
                  "v"((unsigned long long)(uintptr_t)g)
               : "memory");
}
__device__ __forceinline__ void wait_asynccnt0() {
  asm volatile("s_wait_asynccnt 0" ::: "memory");
}

// ---------------------------------------------------------------------------
__global__ void k_init(int* __restrict__ hist, int* __restrict__ meta,
                       float* __restrict__ cscores, float* __restrict__ cboxes,
                       int B) {
  int g = blockIdx.x * blockDim.x + threadIdx.x;
  int stride = gridDim.x * blockDim.x;
  for (int i = g; i < B * NBINS; i += stride) hist[i] = 0;
  for (int i = g; i < B * 4;     i += stride) meta[i] = 0;
  for (int i = g; i < B * PRE;   i += stride) cscores[i] = NEG_INF;
  for (int i = g; i < B * PRE * 4; i += stride) cboxes[i] = 0.0f;
}

// ---------------------------------------------------------------------------
// Softmax over K=9 contiguous logits per (b, fh, fw) cell.
__global__ void k_softmax(const float* __restrict__ labels,
                          float* __restrict__ scores,
                          int total_cells /* B*FH*FW */) {
  int g = blockIdx.x * blockDim.x + threadIdx.x;
  if (g >= total_cells) return;
  const float* in = labels + (size_t)g * KANCH;
  float m = in[0];
#pragma unroll
  for (int k = 1; k < KANCH; ++k) m = fmaxf(m, in[k]);
  float e[KANCH];
  float s = 0.0f;
#pragma unroll
  for (int k = 0; k < KANCH; ++k) { e[k] = __expf(in[k] - m); s += e[k]; }
  float inv = 1.0f / s;
  float* out = scores + (size_t)g * KANCH;
#pragma unroll
  for (int k = 0; k < KANCH; ++k) out[k] = e[k] * inv;
}

// ---------------------------------------------------------------------------
// Per-image score histogram; softmax scores lie in [0,1).
__global__ void k_hist(const float* __restrict__ scores,
                       int* __restrict__ hist, int A) {
  __shared__ int lh[NBINS];
  const int b = blockIdx.y;
  for (int i = threadIdx.x; i < NBINS; i += blockDim.x) lh[i] = 0;
  __syncthreads();
  const float* s = scores + (size_t)b * A;
  int stride = gridDim.x * blockDim.x;
  for (int i = blockIdx.x * blockDim.x + threadIdx.x; i < A; i += stride) {
    int bin = min(NBINS - 1, (int)(s[i] * (float)NBINS));
    atomicAdd(&lh[bin], 1);                       // ds_add_u32
  }
  __syncthreads();
  int* gh = hist + (size_t)b * NBINS;
  for (int i = threadIdx.x; i < NBINS; i += blockDim.x) {
    int v = lh[i];
    if (v) atomicAdd(&gh[i], v);
  }
}

// ---------------------------------------------------------------------------
// Find threshold bin T (all bins > T taken) and boundary budget from bin T
// so exactly PRE candidates are selected. meta[b*4+{0,1}] are counters,
// {2,3} are T and need.
__global__ void k_thresh(const int* __restrict__ hist, int* __restrict__ meta) {
  if (threadIdx.x != 0) return;
  const int b = blockIdx.x;
  const int* h = hist + (size_t)b * NBINS;
  int cum = 0, T = 0, need = PRE;
  for (int t = NBINS - 1; t >= 0; --t) {
    int c = h[t];
    if (cum + c >= PRE) { T = t; need = PRE - cum; break; }
    cum += c;
  }
  meta[b * 4 + 2] = T;
  meta[b * 4 + 3] = need;
}

// ---------------------------------------------------------------------------
// Compact top-6000 per image; decode boxes only for selected anchors.
// anchors layout [y1,x1,y2,x2]; deltas [dy,dx,dh,dw] * var [.1,.1,.2,.2].
__global__ void k_compact(const float* __restrict__ scores,
                          const float* __restrict__ deltas,
                          const float* __restrict__ anchors,
                          int* __restrict__ meta,
                          float* __restrict__ cscores,
                          float* __restrict__ cboxes, int A) {
  const int b = blockIdx.y;
  const float*  s   = scores + (size_t)b * A;
  const float4* dlt = (const float4*)deltas + (size_t)b * A;
  const float4* anc = (const float4*)anchors;
  const int T    = meta[b * 4 + 2];
  const int need = meta[b * 4 + 3];
  int stride = gridDim.x * blockDim.x;
  for (int i = blockIdx.x * blockDim.x + threadIdx.x; i < A; i += stride) {
    float sc = s[i];
    int bin = min(NBINS - 1, (int)(sc * (float)NBINS));
    if (bin < T) continue;
    int slot = -1;
    if (bin > T) {
      slot = atomicAdd(&meta[b * 4 + 0], 1);
    } else {                                  // boundary bin: gated budget
      int t = atomicAdd(&meta[b * 4 + 1], 1);
      if (t < need) slot = atomicAdd(&meta[b * 4 + 0], 1);
    }
    if (slot < 0 || slot >= PRE) continue;
    float4 a = anc[i];
    float4 d = dlt[i];
    float aw = a.w - a.y;
    float ah = a.z - a.x;
    float cx = a.y + 0.5f * aw;
    float cy = a.x + 0.5f * ah;
    float dy = d.x * 0.1f, dx = d.y * 0.1f;
    float dh = d.z * 0.2f, dw = d.w * 0.2f;
    float bw  = __expf(dw) * aw;
    float bh  = __expf(dh) * ah;
    float bcx = dx * aw + cx;
    float bcy = dy * ah + cy;
    float y1 = bcy - 0.5f * bh;
    float x1 = bcx - 0.5f * bw;
    float y2 = bh + y1;
    float x2 = bw + x1;
    ((float4*)cboxes)[(size_t)b * PRE + slot] = make_float4(y1, x1, y2, x2);
    cscores[(size_t)b * PRE + slot] = sc;
  }
}

// ---------------------------------------------------------------------------
// NMS: one workgroup (1024 threads = 32 wave32s) per image.
// LDS: scores 24000B | boxes(AoS float4) 96000B | areas 24000B | reduce 512B.
// Total 144512B — only feasible thanks to CDNA5's 320KB-per-WGP LDS.
static constexpr int NMS_SMEM = PRE * 4 + PRE * 16 + PRE * 4 + 512;

__global__ __launch_bounds__(1024) void k_nms(const float* __restrict__ cscores,
                                              const float* __restrict__ cboxes,
                                              float* __restrict__ out_boxes,
                                              float* __restrict__ out_scores) {
  extern __shared__ unsigned char smem[];
  float*  sS = (float*)smem;                              // [PRE]
  float4* sB = (float4*)(smem + PRE * 4);                 // [PRE]
  float*  sA = (float*)(smem + PRE * 4 + PRE * 16);       // [PRE]
  float*  rV = (float*)(smem + PRE * 24);                 // [64]
  int*    rI = (int*)(smem + PRE * 24 + 256);             // [64]

  const int b   = blockIdx.x;
  const int tid = threadIdx.x;
  const int nthreads = blockDim.x;

  // --- async DMA fill of LDS (ASYNCcnt path) ---
  const float* cs = cscores + (size_t)b * PRE;
  const float* cb = cboxes  + (size_t)b * PRE * 4;
  for (int t = tid; t < PRE / 4; t += nthreads)           // 1500 x 16B scores
    async_b128_to_lds((unsigned)(uintptr_t)(sS + t * 4), cs + t * 4);
  for (int t = tid; t < PRE; t += nthreads)               // 6000 x 16B boxes
    async_b128_to_lds((unsigned)(uintptr_t)(sB + t), cb + t * 4);
  wait_asynccnt0();
  __syncthreads();

  for (int j = tid; j < PRE; j += nthreads) {
    float4 bb = sB[j];
    sA[j] = (bb.z - bb.x) * (bb.w - bb.y);
  }
  __syncthreads();

  float* ob = out_boxes  + (size_t)b * POST * 4;
  float* os = out_scores + (size_t)b * POST;

  for (int it = 0; it < POST; ++it) {
    // ---- parallel argmax (lowest-index tie-break, matches jnp.argmax) ----
    float v = NEG_INF;
    int   ix = 0x7fffffff;
    for (int j = tid; j < PRE; j += nthreads) {
      float s = sS[j];
      if (s > v || (s == v && j < ix)) { v = s; ix = j; }
    }
    for (int off = 16; off > 0; off >>= 1) {              // wave32 tree
      float ov = __shfl_down(v, off, 32);
      int   oi = __shfl_down(ix, off, 32);
      if (ov > v || (ov == v && oi < ix)) { v = ov; ix = oi; }
    }
    const int wave = tid >> 5;
    if ((tid & 31) == 0) { rV[wave] = v; rI[wave] = ix; }
    __syncthreads();
    if (tid < 32) {
      const int nw = nthreads >> 5;
      v  = (tid < nw) ? rV[tid] : NEG_INF;
      ix = (tid < nw) ? rI[tid] : 0x7fffffff;
      for (int off = 16; off > 0; off >>= 1) {
        float ov = __shfl_down(v, off, 32);
        int   oi = __shfl_down(ix, off, 32);
        if (ov > v || (ov == v && oi < ix)) { v = ov; ix = oi; }
      }
      if (tid == 0) { rV[32] = v; rI[32] = (ix == 0x7fffffff) ? 0 : ix; }
    }
    __syncthreads();

    const float bestV = rV[32];
    const int   bestI = rI[32];
    const bool  valid = bestV > NEG_INF;
    const float4 bb = sB[bestI];

    if (tid == 0) {
      float f = valid ? 1.0f : 0.0f;
      ob[it * 4 + 0] = bb.x * f;
      ob[it * 4 + 1] = bb.y * f;
      ob[it * 4 + 2] = bb.z * f;
      ob[it * 4 + 3] = bb.w * f;
      os[it] = valid ? bestV : 0.0f;
    }

    // ---- suppress (chosen box self-suppresses via IoU == 1) ----
    if (valid) {
      const float by1 = bb.x, bx1 = bb.y, by2 = bb.z, bx2 = bb.w;
      const float barea = (by2 - by1) * (bx2 - bx1);
      for (int j = tid; j < PRE; j += nthreads) {
        if (sS[j] == NEG_INF) continue;
        float4 c = sB[j];
        float iy1 = fmaxf(by1, c.x), ix1 = fmaxf(bx1, c.y);
        float iy2 = fminf(by2, c.z), ix2 = fminf(bx2, c.w);
        float inter = fmaxf(iy2 - iy1, 0.0f) * fmaxf(ix2 - ix1, 0.0f);
        float iou = inter / (sA[j] + barea - inter + 1e-9f);
        if (iou > IOU_T) sS[j] = NEG_INF;
      }
    }
    __syncthreads();
  }
}

// ---------------------------------------------------------------------------
extern "C" void kernel_launch(void* const* d_in, const int* in_sizes, int n_in,
                              void* d_out, int out_size, void* d_ws, size_t ws_size,
                              hipStream_t stream) {
  const float* deltas  = (const float*)d_in[0];   // (B, A, 4)
  const float* labels  = (const float*)d_in[1];   // (B, FH, FW, K)
  const float* anchors = (const float*)d_in[2];   // (A, 4)

  const int A = in_sizes[2] / 4;                  // 147456
  const int B = in_sizes[0] / (A * 4);            // 32
  const int ncells = A / KANCH;                   // 16384 per image

  // Carve workspace (~24 MB total), 256B-aligned regions.
  unsigned char* ws = (unsigned char*)d_ws;
  size_t off = 0;
  auto carve = [&](size_t bytes) -> void* {
    void* p = ws + off;
    off = (off + bytes + 255) & ~(size_t)255;
    return p;
  };
  float* scores  = (float*)carve((size_t)B * A * sizeof(float));
  int*   hist    = (int*)  carve((size_t)B * NBINS * sizeof(int));
  int*   meta    = (int*)  carve((size_t)B * 4 * sizeof(int));
  float* cscores = (float*)carve((size_t)B * PRE * sizeof(float));
  float* cboxes  = (float*)carve((size_t)B * PRE * 4 * sizeof(float));
  (void)ws_size; (void)n_in;

  float* out_boxes  = (float*)d_out;              // (B, 300, 4)
  float* out_scores = out_boxes + (size_t)B * POST * 4;

  k_init<<<512, 256, 0, stream>>>(hist, meta, cscores, cboxes, B);

  int total_cells = B * ncells;
  k_softmax<<<(total_cells + 255) / 256, 256, 0, stream>>>(labels, scores, total_cells);

  k_hist<<<dim3(32, B), 256, 0, stream>>>(scores, hist, A);

  k_thresh<<<B, 32, 0, stream>>>(hist, meta);

  k_compact<<<dim3(48, B), 256, 0, stream>>>(scores, deltas, anchors, meta,
                                             cscores, cboxes, A);

  k_nms<<<B, 1024, NMS_SMEM, stream>>>(cscores, cboxes, out_boxes, out_scores);
}